// DynamicHyperGNN_34883724378625
// MI455X (gfx1250) — compile-verified
//
#include <hip/hip_runtime.h>

// ---------------------------------------------------------------------------
// DynamicHyperGNN for MI455X (gfx1250, wave32).
// Memory-bound: ~6.6 GB of gather/scatter traffic dominates (~0.3 ms @ 23.3 TB/s);
// the 13 GFLOP of GEMM goes through the f32 WMMA tensor path (16x16x4) to keep
// exact f32 numerics while using the CDNA5 matrix pipe.
// ---------------------------------------------------------------------------

typedef float v2f __attribute__((ext_vector_type(2)));
typedef float v8f __attribute__((ext_vector_type(8)));

#define DMODEL 256
#define INDIM  240
#define TDIM   16
#define KEDGE  16

// ---- stage 0: max/sum reduction over visit_times -------------------------
__global__ void time_reduce_kernel(const float* __restrict__ t, int E,
                                   unsigned* __restrict__ maxbits,
                                   float* __restrict__ sum) {
  __shared__ float smax[256];
  __shared__ float ssum[256];
  int tid = threadIdx.x;
  float lm = 0.0f, ls = 0.0f;            // times are >= 0
  for (int i = blockIdx.x * blockDim.x + tid; i < E; i += gridDim.x * blockDim.x) {
    float v = t[i];
    lm = fmaxf(lm, v);
    ls += v;
  }
  smax[tid] = lm; ssum[tid] = ls;
  __syncthreads();
  for (int off = 128; off > 0; off >>= 1) {
    if (tid < off) {
      smax[tid] = fmaxf(smax[tid], smax[tid + off]);
      ssum[tid] += ssum[tid + off];
    }
    __syncthreads();
  }
  if (tid == 0) {
    atomicMax(maxbits, __float_as_uint(smax[0]));  // valid: non-negative floats
    atomicAdd(sum, ssum[0]);
  }
}

// ---- per-edge decay weights + broadcast time embedding -------------------
// w[e] = exp((t[e]-maxt)/365);  tnode[j] = mean(t)*Wt[j] + bt[j]
__global__ void edge_weights_kernel(const float* __restrict__ t, int E,
                                    const float* __restrict__ scal,
                                    const float* __restrict__ Wt,
                                    const float* __restrict__ bt,
                                    float* __restrict__ w,
                                    float* __restrict__ tnode) {
  int i = blockIdx.x * blockDim.x + threadIdx.x;
  float maxt = __uint_as_float(((const unsigned*)scal)[0]);
  if (i < E) w[i] = __expf((t[i] - maxt) * (1.0f / 365.0f));
  if (i < TDIM) tnode[i] = (scal[1] / (float)E) * Wt[i] + bt[i];
}

// ---- node degrees dv[n] = sum_incidence w[e];  cnt[n] = #incidences ------
__global__ void incidence_kernel(const int* __restrict__ edges,
                                 const float* __restrict__ w, int total,
                                 float* __restrict__ dv, float* __restrict__ cnt) {
  int i = blockIdx.x * blockDim.x + threadIdx.x;
  if (i < total) {
    int e = i >> 4;                       // KEDGE == 16
    int node = edges[i];
    atomicAdd(&dv[node], w[e]);
    atomicAdd(&cnt[node], 1.0f);
  }
}

__global__ void norm_kernel(const float* __restrict__ dv, const float* __restrict__ cnt,
                            int N, float* __restrict__ dvi, float* __restrict__ ci) {
  int n = blockIdx.x * blockDim.x + threadIdx.x;
  if (n < N) {
    float d = dv[n];
    dvi[n] = d > 0.0f ? rsqrtf(fmaxf(d, 1e-12f)) : 0.0f;
    ci[n]  = 1.0f / fmaxf(cnt[n], 1.0f);
  }
}

// ---- H0 = concat(X, tnode) ----------------------------------------------
__global__ void concat_kernel(const float* __restrict__ X,
                              const float* __restrict__ tnode,
                              int N, float* __restrict__ H) {
  size_t i = (size_t)blockIdx.x * blockDim.x + threadIdx.x;
  int n = (int)(i >> 8);
  int d = (int)(i & 255);
  if (n < N) H[i] = (d < INDIM) ? X[(size_t)n * INDIM + d] : tnode[d - INDIM];
}

// ---- GEMM: Z = H[N x 256] @ W[256 x 256] + b, via V_WMMA_F32_16X16X4_F32 -
// One wave per 16x16 output tile; 64 WMMA steps over K=256.
// A layout (16x4 f32): lane m = lane&15, vgpr j holds K = 2*(lane>>4)+j.
// B layout (4x16 f32): lane n = lane&15, vgpr j holds K = 2*(lane>>4)+j.
// C/D layout: vgpr v -> row M = v + 8*(lane>>4), col N = lane&15.
__global__ void gemm_wmma_kernel(const float* __restrict__ H,
                                 const float* __restrict__ W,
                                 const float* __restrict__ bias,
                                 int Nrows, float* __restrict__ Z) {
  const int gwave = (blockIdx.x * blockDim.x + threadIdx.x) >> 5;
  const int lane  = threadIdx.x & 31;
  const int mt = gwave >> 4;             // DMODEL/16 == 16 column tiles
  const int nt = gwave & 15;
  const int mtiles = (Nrows + 15) >> 4;
  if (mt >= mtiles) return;              // wave-uniform: EXEC stays all-1 for WMMA
  const int l16  = lane & 15;
  const int half = lane >> 4;
  int mrow = mt * 16 + l16;
  if (mrow >= Nrows) mrow = Nrows - 1;   // clamp (N=50000 is an exact multiple anyway)
  const int ncol = nt * 16 + l16;
  const float* Arow = H + (size_t)mrow * DMODEL;

  v8f acc = {};
  for (int k0 = 0; k0 < DMODEL; k0 += 4) {
    const int ka = k0 + half * 2;
    v2f a = *(const v2f*)(Arow + ka);                      // 8B-aligned
    v2f b;
    b.x = W[(size_t)(ka + 0) * DMODEL + ncol];
    b.y = W[(size_t)(ka + 1) * DMODEL + ncol];
    acc = __builtin_amdgcn_wmma_f32_16x16x4_f32(
        /*neg_a=*/false, a, /*neg_b=*/false, b,
        /*c_mod=*/(short)0, acc, /*reuse_a=*/false, /*reuse_b=*/false);
  }
  const float bn = bias[ncol];
#pragma unroll
  for (int v = 0; v < 8; ++v) {
    int row = mt * 16 + v + half * 8;
    if (row < Nrows) Z[(size_t)row * DMODEL + ncol] = acc[v] + bn;
  }
}

// ---- smoothing gather: msg[e] = (w[e]/K) * sum_j Z[nd_j] * dv_isqrt[nd_j]
__global__ void smooth_gather_kernel(const float* __restrict__ Z,
                                     const int* __restrict__ edges,
                                     const float* __restrict__ w,
                                     const float* __restrict__ dvi,
                                     float* __restrict__ msg) {
  const int e = blockIdx.x;
  const int d = threadIdx.x;             // 256 threads = feature dim
  __shared__ int   nd[KEDGE];
  __shared__ float sc[KEDGE];
  if (d < KEDGE) {
    int nid = edges[e * KEDGE + d];
    nd[d] = nid;
    sc[d] = dvi[nid];
  }
  __syncthreads();
  float s = 0.0f;
#pragma unroll
  for (int j = 0; j < KEDGE; ++j)
    s += Z[(size_t)nd[j] * DMODEL + d] * sc[j];
  msg[(size_t)e * DMODEL + d] = s * (w[e] * (1.0f / (float)KEDGE));
}

// ---- scatter-add: out[node] += msg[e] for each incidence ----------------
__global__ void scatter_add_kernel(const float* __restrict__ msg,
                                   const int* __restrict__ edges,
                                   float* __restrict__ out) {
  const int e = blockIdx.x;
  const int d = threadIdx.x;
  __shared__ int nd[KEDGE];
  if (d < KEDGE) nd[d] = edges[e * KEDGE + d];
  __syncthreads();
  float v = msg[(size_t)e * DMODEL + d];
#pragma unroll
  for (int j = 0; j < KEDGE; ++j)
    atomicAdd(&out[(size_t)nd[j] * DMODEL + d], v);
}

// ---- attention gather: ef = mean_j S[nd_j]*dv_isqrt[nd_j] (folds trailing
//      dv_isqrt of smoothing); a = sigmoid(ef.aw + ab); msg = a*ef ---------
__global__ void att_gather_kernel(const float* __restrict__ S,
                                  const int* __restrict__ edges,
                                  const float* __restrict__ dvi,
                                  const float* __restrict__ aw,
                                  const float* __restrict__ ab,
                                  float* __restrict__ msg) {
  const int e = blockIdx.x;
  const int d = threadIdx.x;
  __shared__ int   nd[KEDGE];
  __shared__ float sc[KEDGE];
  __shared__ float red[256];
  if (d < KEDGE) {
    int nid = edges[e * KEDGE + d];
    nd[d] = nid;
    sc[d] = dvi[nid];
  }
  __syncthreads();
  float s = 0.0f;
#pragma unroll
  for (int j = 0; j < KEDGE; ++j)
    s += S[(size_t)nd[j] * DMODEL + d] * sc[j];
  const float ef = s * (1.0f / (float)KEDGE);
  red[d] = ef * aw[d];
  __syncthreads();
  for (int off = 128; off > 0; off >>= 1) {
    if (d < off) red[d] += red[d + off];
    __syncthreads();
  }
  const float att = 1.0f / (1.0f + __expf(-(red[0] + ab[0])));
  msg[(size_t)e * DMODEL + d] = att * ef;
}

// ---- finalize: h = leaky_relu(acc / max(cnt,1), 0.2) --------------------
__global__ void finalize_kernel(const float* __restrict__ acc,
                                const float* __restrict__ ci,
                                int N, float* __restrict__ out) {
  size_t i = (size_t)blockIdx.x * blockDim.x + threadIdx.x;
  int n = (int)(i >> 8);
  if (n < N) {
    float v = acc[i] * ci[n];
    out[i] = v >= 0.0f ? v : 0.2f * v;
  }
}

// ---------------------------------------------------------------------------
extern "C" void kernel_launch(void* const* d_in, const int* in_sizes, int n_in,
                              void* d_out, int out_size, void* d_ws, size_t ws_size,
                              hipStream_t stream) {
  (void)n_in; (void)out_size; (void)ws_size;
  const float* X     = (const float*)d_in[0];
  const int*   edges = (const int*)d_in[1];
  const float* vt    = (const float*)d_in[2];
  const float* Wt    = (const float*)d_in[3];
  const float* bt    = (const float*)d_in[4];
  const float* theta[2] = {(const float*)d_in[5], (const float*)d_in[9]};
  const float* bvec[2]  = {(const float*)d_in[6], (const float*)d_in[10]};
  const float* aw[2]    = {(const float*)d_in[7], (const float*)d_in[11]};
  const float* ab[2]    = {(const float*)d_in[8], (const float*)d_in[12]};
  float* out = (float*)d_out;

  const int N = in_sizes[0] / INDIM;     // 50000
  const int E = in_sizes[2];             // 50000

  // ---- workspace layout (all 256B aligned) ----
  char* ws = (char*)d_ws;
  auto al = [](size_t x) { return (x + 255) & ~(size_t)255; };
  size_t off = 0;
  float* scal = (float*)(ws + off); off = al(off + 256);               // [0]=max bits,[1]=sum,[8..23]=tnode
  float* w    = (float*)(ws + off); off = al(off + (size_t)E * 4);
  float* dv   = (float*)(ws + off); off = al(off + (size_t)N * 4);
  float* cnt  = (float*)(ws + off); off = al(off + (size_t)N * 4);
  float* dvi  = (float*)(ws + off); off = al(off + (size_t)N * 4);
  float* ci   = (float*)(ws + off); off = al(off + (size_t)N * 4);
  const size_t zero_bytes = off;                                       // covers scal/dv/cnt
  float* bufA = (float*)(ws + off); off = al(off + (size_t)N * DMODEL * 4);  // H (layer in/out)
  float* bufB = (float*)(ws + off); off = al(off + (size_t)N * DMODEL * 4);  // Z / att accum
  float* bufC = (float*)(ws + off); off = al(off + (size_t)E * DMODEL * 4);  // edge msgs
  // d_out doubles as the smoothing accumulator S.

  hipMemsetAsync(ws, 0, zero_bytes, stream);

  time_reduce_kernel<<<256, 256, 0, stream>>>(vt, E, (unsigned*)scal, scal + 1);
  edge_weights_kernel<<<(E + 255) / 256, 256, 0, stream>>>(vt, E, scal, Wt, bt, w, scal + 8);
  incidence_kernel<<<(E * KEDGE + 255) / 256, 256, 0, stream>>>(edges, w, E * KEDGE, dv, cnt);
  norm_kernel<<<(N + 255) / 256, 256, 0, stream>>>(dv, cnt, N, dvi, ci);
  concat_kernel<<<((unsigned)N * DMODEL + 255) / 256, 256, 0, stream>>>(X, scal + 8, N, bufA);

  for (int layer = 0; layer < 2; ++layer) {
    const int mtiles = (N + 15) / 16;
    const int waves  = mtiles * (DMODEL / 16);
    const int blocks = (waves + 7) / 8;                    // 8 waves / 256-thread block
    gemm_wmma_kernel<<<blocks, 256, 0, stream>>>(bufA, theta[layer], bvec[layer], N, bufB);

    smooth_gather_kernel<<<E, 256, 0, stream>>>(bufB, edges, w, dvi, bufC);
    hipMemsetAsync(out, 0, (size_t)N * DMODEL * 4, stream);
    scatter_add_kernel<<<E, 256, 0, stream>>>(bufC, edges, out);

    att_gather_kernel<<<E, 256, 0, stream>>>(out, edges, dvi, aw[layer], ab[layer], bufC);
    hipMemsetAsync(bufB, 0, (size_t)N * DMODEL * 4, stream);
    scatter_add_kernel<<<E, 256, 0, stream>>>(bufC, edges, bufB);

    finalize_kernel<<<((unsigned)N * DMODEL + 255) / 256, 256, 0, stream>>>(
        bufB, ci, N, layer == 0 ? bufA : out);
  }
}